// CRFLoss_ma_mturk_71631464563257
// MI455X (gfx1250) — compile-verified
//
#include <hip/hip_runtime.h>
#include <stdint.h>

// ---------------------------------------------------------------------------
// CRF loss for MI455X (gfx1250).
//  Phase 1: chunked log-semiring matrix products (WMMA f32 16x16x4 in exp
//           domain, per-row-max stabilized), 4096 blocks -> bandwidth bound.
//  Phase 2: 64-step scan of partition vector over chunk matrices (broadcast-A
//           WMMA matvec).
//  Async global->LDS pipeline (ASYNCcnt) for all tile streaming.
// ---------------------------------------------------------------------------

#define Lx 512
#define Bx 64
#define Tx 64
#define TTx 4096
#define NCHUNK 64
#define CH 8
#define NSTEPS 511   /* l = 1..511 */

#define AS1 __attribute__((address_space(1)))
#define AS3 __attribute__((address_space(3)))

#ifndef __has_builtin
#define __has_builtin(x) 0
#endif

#if __has_builtin(__builtin_amdgcn_global_load_async_to_lds_b128) && \
    __has_builtin(__builtin_amdgcn_s_wait_asynccnt)
#define HAVE_ASYNC 1
#else
#define HAVE_ASYNC 0
#endif

#if __has_builtin(__builtin_amdgcn_wmma_f32_16x16x4_f32)
#define HAVE_WMMA_F32 1
#else
#define HAVE_WMMA_F32 0
#endif

typedef float v2f __attribute__((ext_vector_type(2)));
typedef float v8f __attribute__((ext_vector_type(8)));
typedef int v4i __attribute__((__vector_size__(16)));   // matches builtin param

// Copy one 16KB (64x64 f32) tile, 128 threads, 8 x b128 async ops per wave.
__device__ __forceinline__ void async_copy_tile(const float* __restrict__ g,
                                                float* l, int tid) {
#if HAVE_ASYNC
#pragma unroll
  for (int r = 0; r < 8; ++r) {
    const int off = (r * 128 + tid) * 4;
    __builtin_amdgcn_global_load_async_to_lds_b128(
        (AS1 v4i*)(float*)(g + off), (AS3 v4i*)(l + off), 0, 0);
  }
#else
  (void)g; (void)l; (void)tid;
#endif
}

__device__ __forceinline__ void sync_copy_tile(const float* __restrict__ g,
                                               float* l, int tid) {
  const float4* g4 = (const float4*)g;
  float4* l4 = (float4*)l;
#pragma unroll
  for (int r = 0; r < 8; ++r) l4[r * 128 + tid] = g4[r * 128 + tid];
}

// ---------------------------------------------------------------------------
// Phase 1: chunk product.  P <- P (logmul) M_l  for 8 consecutive tiles.
// grid = (Bx, NCHUNK), block = 128 (4 waves).
// ---------------------------------------------------------------------------
__global__ __launch_bounds__(128)
void crf_chunk_kernel(const float* __restrict__ scores,
                      const unsigned char* __restrict__ mask,
                      float* __restrict__ chunkP) {
  const int b = blockIdx.x;
  const int c = blockIdx.y;
  const int tid = threadIdx.x;
  const int lane = tid & 31;
  const int wave = tid >> 5;
  const int s0 = c * CH;
  const int ns = (CH < NSTEPS - s0) ? CH : (NSTEPS - s0);

  __shared__ __align__(16) float P[TTx];       // 16 KB: log-P, then exp in place
  __shared__ __align__(16) float tile[2][TTx]; // 32 KB: double buffer
  __shared__ float rmax[Tx];

  // log-domain identity
  for (int k = tid; k < TTx; k += 128) {
    const int i = k >> 6, j = k & 63;
    P[k] = (i == j) ? 0.0f : -1e30f;
  }

  const float* gbase = scores + ((size_t)(1 + s0) * Bx + b) * (size_t)TTx;
  const size_t gstride = (size_t)Bx * TTx;

#if HAVE_ASYNC
  for (int k = 0; k < 2 && k < ns; ++k)
    async_copy_tile(gbase + (size_t)k * gstride, tile[k], tid);
#endif

  for (int s = 0; s < ns; ++s) {
    float* Mt = tile[s & 1];
#if HAVE_ASYNC
    if (s + 2 <= ns) __builtin_amdgcn_s_wait_asynccnt(8);  // own 8 ops for tile s done
    else             __builtin_amdgcn_s_wait_asynccnt(0);
    __syncthreads();
#else
    __syncthreads();
    sync_copy_tile(gbase + (size_t)s * gstride, Mt, tid);
    __syncthreads();
#endif
    const int mv = mask[(size_t)(1 + s0 + s) * Bx + b];
    if (mv) {
      // per-row max of log-P (finite: diagonal >= 0-ish always present)
      if (tid < 64) {
        float m = P[tid * 64];
        for (int k = 1; k < 64; ++k) m = fmaxf(m, P[tid * 64 + k]);
        rmax[tid] = m;
      }
      __syncthreads();
      // exp-transform P in place: EP[i][k] = exp(P - rmax[i]) in [0,1]
      for (int k = tid; k < TTx; k += 128) P[k] = __expf(P[k] - rmax[k >> 6]);
      __syncthreads();

#if HAVE_WMMA_F32
      // wave -> column block jb, 4 row tiles, 16 k-chunks of WMMA f32 16x16x4.
      const int jb = wave * 16;
      const int h = lane >> 4;   // K-half select per documented layouts
      const int n = lane & 15;   // A row / B,D column within tile
      v8f acc0 = {}, acc1 = {}, acc2 = {}, acc3 = {};
#pragma unroll
      for (int kc = 0; kc < 16; ++kc) {
        const int k0 = kc * 4 + 2 * h;
        v2f bf, a0, a1, a2, a3;
        bf[0] = __expf(Mt[k0 * 64 + jb + n]);        // exp(M), unshifted (|M|~6)
        bf[1] = __expf(Mt[(k0 + 1) * 64 + jb + n]);
        a0[0] = P[(n)      * 64 + k0]; a0[1] = P[(n)      * 64 + k0 + 1];
        a1[0] = P[(16 + n) * 64 + k0]; a1[1] = P[(16 + n) * 64 + k0 + 1];
        a2[0] = P[(32 + n) * 64 + k0]; a2[1] = P[(32 + n) * 64 + k0 + 1];
        a3[0] = P[(48 + n) * 64 + k0]; a3[1] = P[(48 + n) * 64 + k0 + 1];
        acc0 = __builtin_amdgcn_wmma_f32_16x16x4_f32(false, a0, false, bf, (short)0, acc0, false, false);
        acc1 = __builtin_amdgcn_wmma_f32_16x16x4_f32(false, a1, false, bf, (short)0, acc1, false, false);
        acc2 = __builtin_amdgcn_wmma_f32_16x16x4_f32(false, a2, false, bf, (short)0, acc2, false, false);
        acc3 = __builtin_amdgcn_wmma_f32_16x16x4_f32(false, a3, false, bf, (short)0, acc3, false, false);
      }
      __syncthreads();                       // all EP/tile reads complete
#if HAVE_ASYNC
      if (s + 2 < ns) async_copy_tile(gbase + (size_t)(s + 2) * gstride, Mt, tid);
#endif
      // D layout: vgpr r -> rows r (lanes 0-15) and r+8 (lanes 16-31)
#pragma unroll
      for (int r = 0; r < 8; ++r) {
        const int ib = r + 8 * h;
        P[(ib)      * 64 + jb + n] = rmax[ib]      + __logf(acc0[r]);
        P[(16 + ib) * 64 + jb + n] = rmax[16 + ib] + __logf(acc1[r]);
        P[(32 + ib) * 64 + jb + n] = rmax[32 + ib] + __logf(acc2[r]);
        P[(48 + ib) * 64 + jb + n] = rmax[48 + ib] + __logf(acc3[r]);
      }
#else
      // VALU fallback
      float outv[32];
#pragma unroll
      for (int e = 0; e < 32; ++e) {
        const int k2 = e * 128 + tid;
        const int i = k2 >> 6, j = k2 & 63;
        float sacc = 0.0f;
        for (int k = 0; k < 64; ++k) sacc += P[i * 64 + k] * __expf(Mt[k * 64 + j]);
        outv[e] = rmax[i] + __logf(sacc);
      }
      __syncthreads();
#if HAVE_ASYNC
      if (s + 2 < ns) async_copy_tile(gbase + (size_t)(s + 2) * gstride, Mt, tid);
#endif
#pragma unroll
      for (int e = 0; e < 32; ++e) P[e * 128 + tid] = outv[e];
#endif
    } else {
      __syncthreads();
#if HAVE_ASYNC
      if (s + 2 < ns) async_copy_tile(gbase + (size_t)(s + 2) * gstride, Mt, tid);
#endif
    }
  }

  __syncthreads();
  float* out = chunkP + ((size_t)c * Bx + b) * (size_t)TTx;
  for (int k = tid; k < TTx; k += 128) out[k] = P[k];
}

// ---------------------------------------------------------------------------
// Phase 2 / direct path: p <- p (logmul) M_s, sequential over nsteps matrices.
// grid = Bx, block = 128 (4 waves, one 16-column block each).
// mats layout: step s matrix at mats + (s*Bx + b)*TTx. maskBase may be null.
// ---------------------------------------------------------------------------
__global__ __launch_bounds__(128)
void crf_scan_kernel(const float* __restrict__ p0src,
                     const float* __restrict__ mats,
                     const unsigned char* __restrict__ maskBase,
                     int nsteps,
                     float* __restrict__ Zpart) {
  const int b = blockIdx.x;
  const int tid = threadIdx.x;
  const int lane = tid & 31;
  const int wave = tid >> 5;

  __shared__ __align__(16) float tile[3][TTx]; // 48 KB triple buffer
  __shared__ float p[Tx];

  if (tid < 64) p[tid] = p0src[(size_t)b * TTx + tid];  // scores[0,b,START_TAG,:]

  const float* gbase = mats + (size_t)b * TTx;
  const size_t gstride = (size_t)Bx * TTx;
  const int NBUF = 3;

#if HAVE_ASYNC
  for (int k = 0; k < NBUF && k < nsteps; ++k)
    async_copy_tile(gbase + (size_t)k * gstride, tile[k], tid);
#endif

  for (int s = 0; s < nsteps; ++s) {
    float* Mt = tile[s % 3];
#if HAVE_ASYNC
    if (s + NBUF <= nsteps) __builtin_amdgcn_s_wait_asynccnt(16);
    else                    __builtin_amdgcn_s_wait_asynccnt(0);
    __syncthreads();
#else
    __syncthreads();
    sync_copy_tile(gbase + (size_t)s * gstride, Mt, tid);
    __syncthreads();
#endif
    const int mv = maskBase ? maskBase[(size_t)s * Bx + b] : 1;
    if (mv) {
      // pm = max_i p[i], butterfly reduce within wave (wave32)
      float v = fmaxf(p[lane], p[lane + 32]);
#pragma unroll
      for (int off = 16; off > 0; off >>= 1) v = fmaxf(v, __shfl_xor(v, off, 32));
      const float pm = v;
      const int jb = wave * 16;
      const int h = lane >> 4;
      const int n = lane & 15;
#if HAVE_WMMA_F32
      // Broadcast-A matvec: A[m][k] = exp(p[k]-pm) for all m -> every D row = s[j]
      v8f acc = {};
#pragma unroll
      for (int kc = 0; kc < 16; ++kc) {
        const int k0 = kc * 4 + 2 * h;
        v2f a, bf;
        a[0]  = __expf(p[k0] - pm);
        a[1]  = __expf(p[k0 + 1] - pm);
        bf[0] = __expf(Mt[k0 * 64 + jb + n]);
        bf[1] = __expf(Mt[(k0 + 1) * 64 + jb + n]);
        acc = __builtin_amdgcn_wmma_f32_16x16x4_f32(false, a, false, bf, (short)0, acc, false, false);
      }
      const float pn = pm + __logf(acc[0]);  // row M=0 lives in vgpr0, lanes 0-15
      __syncthreads();
      if (h == 0) p[jb + n] = pn;
#else
      float pn = 0.0f;
      if (tid < 64) {
        float sacc = 0.0f;
        for (int i = 0; i < 64; ++i)
          sacc += __expf(p[i] - pm) * __expf(Mt[i * 64 + tid]);
        pn = pm + __logf(sacc);
      }
      __syncthreads();
      if (tid < 64) p[tid] = pn;
#endif
#if HAVE_ASYNC
      if (s + NBUF < nsteps)
        async_copy_tile(gbase + (size_t)(s + NBUF) * gstride, Mt, tid);
#endif
    } else {
      __syncthreads();
#if HAVE_ASYNC
      if (s + NBUF < nsteps)
        async_copy_tile(gbase + (size_t)(s + NBUF) * gstride, Mt, tid);
#endif
    }
  }
  __syncthreads();
  if (tid == 0) Zpart[b] = p[1];  // END_TAG
}

// ---------------------------------------------------------------------------
// tg_energy gather-reduce and finalize
// ---------------------------------------------------------------------------
__global__ void crf_init_kernel(float* accum) { accum[0] = 0.0f; }

__global__ __launch_bounds__(256)
void crf_tg_kernel(const float* __restrict__ scores,
                   const int* __restrict__ target,
                   const unsigned char* __restrict__ mask,
                   float* __restrict__ accum) {
  const int idx = blockIdx.x * 256 + threadIdx.x;  // l*B + b
  float v = 0.0f;
  if (idx < Lx * Bx && mask[idx]) {
    const int t = target[idx];
    v = scores[(size_t)idx * TTx + (size_t)t];
  }
#pragma unroll
  for (int off = 16; off > 0; off >>= 1) v += __shfl_xor(v, off, 32);
  __shared__ float red[8];
  if ((threadIdx.x & 31) == 0) red[threadIdx.x >> 5] = v;
  __syncthreads();
  if (threadIdx.x == 0) {
    float sv = 0.0f;
    for (int w = 0; w < 8; ++w) sv += red[w];
    atomicAdd(accum, sv);
  }
}

__global__ void crf_final_kernel(const float* __restrict__ antor,
                                 const int* __restrict__ aidp,
                                 const float* __restrict__ Zpart,
                                 const float* __restrict__ accum,
                                 float* __restrict__ out) {
  float Z = 0.0f;
  for (int b = 0; b < Bx; ++b) Z += Zpart[b];
  const int aid = aidp[0];  // low 32 bits; value is small & little-endian safe
  float m = antor[0];
  for (int k = 1; k < 8; ++k) m = fmaxf(m, antor[k]);
  float den = 0.0f;
  for (int k = 0; k < 8; ++k) den += __expf(antor[k] - m);
  const float w = __expf(antor[aid] - m) / den;
  out[0] = (Z - accum[0]) * w / (float)Bx;
}

// ---------------------------------------------------------------------------
extern "C" void kernel_launch(void* const* d_in, const int* in_sizes, int n_in,
                              void* d_out, int out_size, void* d_ws, size_t ws_size,
                              hipStream_t stream) {
  (void)in_sizes; (void)n_in; (void)out_size;
  const float* scores        = (const float*)d_in[0];
  const int* target          = (const int*)d_in[1];   // JAX x64-off: int64->int32
  const unsigned char* mask  = (const unsigned char*)d_in[2];
  const float* antor         = (const float*)d_in[3];
  const int* aidp            = (const int*)d_in[4];
  float* outp                = (float*)d_out;

  float* ws     = (float*)d_ws;
  float* accum  = ws;            // [0]
  float* Zpart  = ws + 64;       // [64..127]
  float* chunkP = ws + 256;      // 64*64*4096 floats = 64 MB
  const size_t need = (256 + (size_t)NCHUNK * Bx * TTx) * sizeof(float);

  crf_init_kernel<<<1, 1, 0, stream>>>(accum);
  crf_tg_kernel<<<(Lx * Bx) / 256, 256, 0, stream>>>(scores, target, mask, accum);

  if (ws_size >= need) {
    // Parallel chunk products (4096 blocks), then 64-step scan over chunks.
    crf_chunk_kernel<<<dim3(Bx, NCHUNK), 128, 0, stream>>>(scores, mask, chunkP);
    crf_scan_kernel<<<Bx, 128, 0, stream>>>(scores, chunkP, nullptr, NCHUNK, Zpart);
  } else {
    // Fallback: direct sequential scan over the 511 score tiles.
    crf_scan_kernel<<<Bx, 128, 0, stream>>>(scores,
                                            scores + (size_t)Bx * TTx,
                                            mask + Bx, NSTEPS, Zpart);
  }
  crf_final_kernel<<<1, 1, 0, stream>>>(antor, aidp, Zpart, accum, outp);
}